// Get_gradient_nopadding_46737834115117
// MI455X (gfx1250) — compile-verified
//
#include <hip/hip_runtime.h>

// Sobel gradient magnitude, x: [8, 64, 256, 256] f32 -> same-shape f32.
// Bandwidth-bound (AI ~ 0.8 flop/byte; floor ~11 us at 23.3 TB/s).
// Strategy: TDM (tensor_load_to_lds) stages a 34x256 row tile (tile + halo)
// into LDS with ONE DMA instruction per block; stencil reuse then happens at
// LDS bandwidth; nontemporal output stores keep the write stream out of L2.

#define TILE_H   32
#define IMG_H    256
#define IMG_W    256
#define LDS_ROWS (TILE_H + 2)

typedef __attribute__((ext_vector_type(4))) unsigned int u32x4;
typedef __attribute__((ext_vector_type(8))) int          i32x8;
typedef __attribute__((ext_vector_type(4))) int          i32x4;
typedef __attribute__((ext_vector_type(4))) float        f32x4;

#ifndef __has_builtin
#define __has_builtin(x) 0
#endif

#if __has_builtin(__builtin_amdgcn_tensor_load_to_lds) && \
    __has_builtin(__builtin_amdgcn_s_wait_tensorcnt)
#define USE_TDM 1
#else
#define USE_TDM 0
#endif

__global__ __launch_bounds__(256) void sobel_mag_kernel(const float* __restrict__ x,
                                                        float* __restrict__ out) {
    __shared__ float sh[LDS_ROWS][IMG_W];   // 34 KB

    const int tid  = threadIdx.x;           // 0..255, lane == column
    const int row0 = blockIdx.x * TILE_H;   // first output row of this tile
    const int bc   = blockIdx.y;            // fused batch*channel index
    const float* img = x + (size_t)bc * (IMG_H * IMG_W);

    // Valid global rows to fetch: [gstart, gend). LDS row lr maps to global
    // row row0 - 1 + lr. Clamp at the image edges.
    const int gstart = (row0 == 0) ? 0 : row0 - 1;
    const int gend   = (row0 + TILE_H + 1 < IMG_H) ? (row0 + TILE_H + 1) : IMG_H;
    const int nrows  = gend - gstart;           // 33 at edges, 34 interior
    const int lr0    = gstart - (row0 - 1);     // 1 if clamped at top else 0

    // Zero-fill halo rows that fall outside the image. These LDS rows are
    // disjoint from the TDM tile, so there is no DS-store vs DMA-write race.
    if (row0 == 0)              sh[0][tid]            = 0.0f;
    if (row0 + TILE_H == IMG_H) sh[LDS_ROWS - 1][tid] = 0.0f;

#if USE_TDM
    // TDM ignores EXEC, so gate at wave granularity: exactly one wave issues
    // the DMA and waits on TENSORcnt; the workgroup barrier below publishes
    // the completed LDS tile to the remaining waves.
    if ((tid >> 5) == 0) {
        const size_t   ga       = (size_t)(img + (size_t)gstart * IMG_W);
        const unsigned lds_addr = (unsigned)(size_t)(&sh[lr0][0]); // low 32b of
                                                                   // generic LDS ptr
        // ---- D# group 0 (128b): count | lds_addr | global_addr | type=2 ----
        u32x4 g0;
        g0[0] = 1u;                                   // count=1 valid user D#
        g0[1] = lds_addr;                             // bits 63:32
        g0[2] = (unsigned)(ga & 0xFFFFFFFFu);         // global_addr[31:0]
        g0[3] = (unsigned)((ga >> 32) & 0x01FFFFFFu)  // global_addr[56:32]
              | (2u << 30);                           // type = 2 ("image")

        // ---- D# group 1 (256b): 2D tensor/tile geometry, fp32 elements ----
        i32x8 g1;
        g1[0] = 0x00020000;                            // data_size=2 (4B), mask=0
        g1[1] = (IMG_W & 0xFFFF) << 16;                // tensor_dim0[15:0] @63:48
        g1[2] = ((IMG_W >> 16) & 0xFFFF)               // tensor_dim0[31:16]
              | ((nrows & 0xFFFF) << 16);              // tensor_dim1[15:0]
        g1[3] = (IMG_W & 0xFFFF) << 16;                // tensor_dim1 hi(0)|tile_dim0
        g1[4] = (nrows & 0xFFFF);                      // tile_dim1 (tile_dim2=0)
        g1[5] = IMG_W;                                 // tensor_dim0_stride lo32
        g1[6] = 0;                                     // stride hi / dim1_stride lo
        g1[7] = 0;

        i32x4 gz4 = {0, 0, 0, 0};                      // groups 2/3: unused (2D)
        i32x8 gz8 = {0, 0, 0, 0, 0, 0, 0, 0};          // extra group (clang-23
                                                       // 6-arg form): zero-filled
        __builtin_amdgcn_tensor_load_to_lds(g0, g1, gz4, gz4, gz8, 0);
        __builtin_amdgcn_s_wait_tensorcnt(0);
    }
#else
    // Fallback: cooperative float4 global->LDS staging with zero halo.
    for (int k = tid; k < LDS_ROWS * (IMG_W / 4); k += 256) {
        const int lr = k >> 6;
        const int c4 = (k & 63) << 2;
        const int g  = row0 - 1 + lr;
        f32x4 v = {0.f, 0.f, 0.f, 0.f};
        if (g >= 0 && g < IMG_H)
            v = *(const f32x4*)(img + (size_t)g * IMG_W + c4);
        *(f32x4*)(&sh[lr][c4]) = v;
    }
#endif
    __syncthreads();

    // Compute: lane owns one column, walks TILE_H rows. LDS accesses are
    // stride-1 across lanes -> conflict-free. Stores are fully coalesced rows.
    const int c = tid;
    float* orow = out + ((size_t)bc * IMG_H + row0) * IMG_W + c;
    #pragma unroll 8
    for (int r = 0; r < TILE_H; ++r) {
        const float gv = sh[r + 2][c] - sh[r][c];
        const float xl = (c > 0)         ? sh[r + 1][c - 1] : 0.0f;
        const float xr = (c < IMG_W - 1) ? sh[r + 1][c + 1] : 0.0f;
        const float gh = xr - xl;
        const float v  = __builtin_sqrtf(gv * gv + gh * gh + 1e-6f);
        __builtin_nontemporal_store(v, orow + (size_t)r * IMG_W);
    }
}

extern "C" void kernel_launch(void* const* d_in, const int* in_sizes, int n_in,
                              void* d_out, int out_size, void* d_ws, size_t ws_size,
                              hipStream_t stream) {
    (void)n_in; (void)out_size; (void)d_ws; (void)ws_size;
    const float* x = (const float*)d_in[0];
    float* out     = (float*)d_out;
    const int bc   = in_sizes[0] / (IMG_H * IMG_W);   // 8 * 64 = 512
    dim3 grid(IMG_H / TILE_H, bc);                    // 8 x 512 blocks
    sobel_mag_kernel<<<grid, 256, 0, stream>>>(x, out);
}